// SplitTableBatchedEmbeddingBagsCodegen_83193516523939
// MI455X (gfx1250) — compile-verified
//
#include <hip/hip_runtime.h>
#include <cstdint>

// Embedding-bag SUM pooling: T=4 tables, E=1M rows, D=128, B=8192, L=32.
// Pure gather-reduce: ~512 MiB random 512-B row reads out of a 2 GiB table
// (doesn't fit 192 MB L2) + 16 MiB output => HBM-bound, ~23 us floor at
// 23.3 TB/s. WMMA gives nothing here; the CDNA5 feature that pays is the
// async global->LDS DMA engine: 32 row copies in flight per wave with near
// zero VGPR pressure, drained in-order via s_wait_asynccnt.

#define T_TABLES 4
#define E_ROWS   1000000
#define D_DIM    128
#define B_BATCH  8192
#define L_BAG    32
#define WAVES_PER_BLOCK 4

typedef float v4f __attribute__((ext_vector_type(4)));

__global__ __launch_bounds__(WAVES_PER_BLOCK * 32)
void embedding_bag_pool_async_kernel(const int* __restrict__ indices,
                                     const float* __restrict__ weights,
                                     float* __restrict__ out) {
    // 4 waves x 32 rows x 512 B = 64 KB per block (of 320 KB/WGP -> 5 blocks).
    __shared__ v4f buf[WAVES_PER_BLOCK][L_BAG * (D_DIM / 4)];

    const int lane = threadIdx.x & 31;
    // Explicitly wave-uniform so bag/t/b/table_base stay in SGPRs.
    const int wave = __builtin_amdgcn_readfirstlane((int)(threadIdx.x >> 5));
    const int bag  = blockIdx.x * WAVES_PER_BLOCK + wave;   // [0, T*B), uniform
    const int t    = bag / B_BATCH;                         // uniform
    const int b    = bag - t * B_BATCH;                     // uniform

    // Uniform address -> scalar (SMEM) loads; rows[] lives in SGPRs.
    const int* __restrict__ bag_idx = indices + (size_t)bag * L_BAG;
    int rows[L_BAG];
#pragma unroll
    for (int i = 0; i < L_BAG; ++i) rows[i] = bag_idx[i];

    const unsigned long long table_base =
        (unsigned long long)(uintptr_t)weights +
        (unsigned long long)t * (unsigned long long)E_ROWS * (D_DIM * 4);

    // Loop-invariant per-lane operands: 16-B chunk offset within a row, and
    // the LDS destination of slot 0. (Low 32 bits of a generic LDS pointer
    // are the hardware LDS byte address.)
    const unsigned lane_byte = (unsigned)(lane * 16);
    const unsigned lds_dst   = (unsigned)(uintptr_t)(&buf[wave][0]) + lane_byte;

    // Issue all 32 row copies (512 B each, 16 B/lane) through the async unit.
    // GVS addressing: saddr64 (SGPR, pure SALU math: row<<9 fits 32 bits)
    // + vaddr32 (lane_byte). LDS slot i selected with one v_add on vdst.
    // ASYNCcnt is 6-bit: 32 outstanding per wave is legal.
#pragma unroll
    for (int i = 0; i < L_BAG; ++i) {
        const unsigned long long srow =
            table_base + (unsigned long long)(((unsigned)rows[i]) << 9);
        asm volatile("global_load_async_to_lds_b128 %0, %1, %2"
                     :: "v"(lds_dst + (unsigned)(i * 512)),
                        "v"(lane_byte),
                        "s"(srow)
                     : "memory");
    }

    // Rolling in-order drain: async loads retire in order, so ASYNCcnt <=
    // (31 - i) guarantees rows 0..i have landed in LDS. Lane reads its own
    // 16-B chunk of each row (ds_load_b128) and accumulates in f32.
    asm volatile("s_wait_asynccnt %0" :: "n"(L_BAG - 1) : "memory");
    v4f acc = buf[wave][lane];
#pragma unroll
    for (int i = 1; i < L_BAG; ++i) {
        asm volatile("s_wait_asynccnt %0" :: "n"(L_BAG - 1 - i) : "memory");
        acc += buf[wave][i * (D_DIM / 4) + lane];
    }

    // out[b, t*D + 4*lane .. +3]: one coalesced 512-B store per wave.
    // Non-temporal: the 16 MiB output shouldn't evict gather lines from L2.
    v4f* dst = (v4f*)(out + (size_t)b * (T_TABLES * D_DIM) +
                      (size_t)t * D_DIM + (size_t)(lane * 4));
    __builtin_nontemporal_store(acc, dst);
}

extern "C" void kernel_launch(void* const* d_in, const int* in_sizes, int n_in,
                              void* d_out, int out_size, void* d_ws, size_t ws_size,
                              hipStream_t stream) {
    (void)in_sizes; (void)n_in; (void)out_size; (void)d_ws; (void)ws_size;
    const int*   indices = (const int*)d_in[0];    // [T, B, L]
    const float* weights = (const float*)d_in[1];  // [T, E, D]
    float*       out     = (float*)d_out;          // [B, T*D]

    const int total_bags = T_TABLES * B_BATCH;                 // 32768
    dim3 grid(total_bags / WAVES_PER_BLOCK);                   // 8192 blocks
    dim3 block(WAVES_PER_BLOCK * 32);                          // 128 threads
    embedding_bag_pool_async_kernel<<<grid, block, 0, stream>>>(indices, weights, out);
}